// Vit_block_59691455480362
// MI455X (gfx1250) — compile-verified
//
#include <hip/hip_runtime.h>
#include <hip/hip_bf16.h>

typedef __attribute__((ext_vector_type(2))) float v2f;
typedef __attribute__((ext_vector_type(8))) float v8f;

#define EMBED 768
#define DIM 800
#define IN_C 1024
#define NPOS 1024   // GRID*GRID
#define BATCH 32
#define SCALE 0.0353553390593273762f   // 1/sqrt(800)

// ---------------------------------------------------------------------------
// Kernel 1: per-batch preparation.
//   t0 = [cls_token(768); gender_b(32)]
//   q0 = Wq * t0 + bq                (rows 0..799 of qkv_w)
//   r  = Wk^T * q0                   (rows 800..1599 of qkv_w)
//   s  = r[:768] * ln_g  -> Sb[b]
//   scal[b] = { sum(s), r[:768].ln_b, r[768:].gender_b, q0.bk }
// ---------------------------------------------------------------------------
__global__ __launch_bounds__(256) void prep_kernel(
    const float* __restrict__ gender, const float* __restrict__ cls,
    const float* __restrict__ qkv_w, const float* __restrict__ qkv_b,
    const float* __restrict__ ln_g, const float* __restrict__ ln_b,
    float* __restrict__ Sb, float* __restrict__ scal)
{
    const int b   = blockIdx.x;
    const int tid = threadIdx.x;
    __shared__ float t0s[DIM];
    __shared__ float q0s[DIM];
    __shared__ float wred[4][8];

    for (int i = tid; i < DIM; i += 256)
        t0s[i] = (i < EMBED) ? cls[i] : gender[b * 32 + (i - EMBED)];
    __syncthreads();

    // q0 rows
    for (int j = tid; j < DIM; j += 256) {
        float acc = qkv_b[j];
        const float* wr = qkv_w + (size_t)j * DIM;
        for (int c = 0; c < DIM; ++c) acc += wr[c] * t0s[c];
        q0s[j] = acc;
    }
    __syncthreads();

    float p_sum_s = 0.f, p_rlnb = 0.f, p_gterm = 0.f, p_cst = 0.f;

    for (int j = tid; j < DIM; j += 256)
        p_cst += q0s[j] * qkv_b[DIM + j];

    // r[c] = sum_j Wk[j][c] * q0[j]  (coalesced across threads in c)
    for (int c = tid; c < DIM; c += 256) {
        float acc = 0.f;
        for (int j = 0; j < DIM; ++j)
            acc += qkv_w[(size_t)(DIM + j) * DIM + c] * q0s[j];
        if (c < EMBED) {
            float sv = acc * ln_g[c];
            Sb[b * EMBED + c] = sv;
            p_sum_s += sv;
            p_rlnb  += acc * ln_b[c];
        } else {
            p_gterm += acc * t0s[c];   // t0s[c] == gender value for c >= 768
        }
    }

    // deterministic reduction: shfl within wave32, then across the 8 waves
    #pragma unroll
    for (int off = 16; off > 0; off >>= 1) {
        p_sum_s += __shfl_xor(p_sum_s, off, 32);
        p_rlnb  += __shfl_xor(p_rlnb,  off, 32);
        p_gterm += __shfl_xor(p_gterm, off, 32);
        p_cst   += __shfl_xor(p_cst,   off, 32);
    }
    const int lane = tid & 31, wave = tid >> 5;
    if (lane == 0) {
        wred[0][wave] = p_sum_s; wred[1][wave] = p_rlnb;
        wred[2][wave] = p_gterm; wred[3][wave] = p_cst;
    }
    __syncthreads();
    if (tid < 4) {
        float acc = 0.f;
        #pragma unroll
        for (int w = 0; w < 8; ++w) acc += wred[tid][w];
        scal[b * 4 + tid] = acc;
    }
}

// ---------------------------------------------------------------------------
// Kernel 2: P[b][n][o] = sum_c x[b][c][n] * conv_w[o][c] + conv_b[o]
// fp32 WMMA 16x16x4. Block tile 128(M=pos) x 64(N=out), K staged by 32 in LDS.
// 8 waves in a 4x2 grid; each wave owns a 32x32 register tile = four 16x16
// accumulators -> 1 LDS load per WMMA and 4 independent accumulate chains.
// ---------------------------------------------------------------------------
__global__ __launch_bounds__(256) void gemm_conv_kernel(
    const float* __restrict__ x, const float* __restrict__ conv_w,
    const float* __restrict__ conv_b, float* __restrict__ P)
{
    const int mt = blockIdx.x;        // 0..7   (1024/128 position tiles)
    const int nt = blockIdx.y;        // 0..11  (768/64 output tiles)
    const int b  = blockIdx.z;        // 0..31
    const int m_base = mt * 128;
    const int o_base = nt * 64;
    const float* xb = x + (size_t)b * IN_C * NPOS;

    __shared__ float As[32][128];     // [k][m]   16 KB
    __shared__ float Bs[32][65];      // [k][n]   padded row vs bank conflicts

    const int tid  = threadIdx.x;
    const int lane = tid & 31;
    const int wave = tid >> 5;
    const int half = lane >> 4;       // 0: lanes 0-15, 1: lanes 16-31
    const int lm   = lane & 15;
    const int mw   = (wave & 3) * 32; // wave's M offset within block tile
    const int nw   = (wave >> 2) * 32;// wave's N offset within block tile

    v8f acc00 = {}, acc01 = {}, acc10 = {}, acc11 = {};

    for (int kb = 0; kb < IN_C; kb += 32) {
        // Stage A tile: 32 channels x 128 positions (positions contiguous in x)
        #pragma unroll
        for (int r = 0; r < 16; ++r) {
            int idx = tid + 256 * r;
            int k = idx >> 7, m = idx & 127;
            As[k][m] = xb[(size_t)(kb + k) * NPOS + m_base + m];
        }
        // Stage B tile: Bs[k][n] = conv_w[(o_base+n)*1024 + kb+k]
        {
            int n  = tid >> 2;            // 0..63
            int kq = (tid & 3) * 8;       // 0,8,16,24
            const float* src = conv_w + (size_t)(o_base + n) * IN_C + kb + kq;
            const float4 v0 = *(const float4*)(src);
            const float4 v1 = *(const float4*)(src + 4);
            Bs[kq + 0][n] = v0.x; Bs[kq + 1][n] = v0.y;
            Bs[kq + 2][n] = v0.z; Bs[kq + 3][n] = v0.w;
            Bs[kq + 4][n] = v1.x; Bs[kq + 5][n] = v1.y;
            Bs[kq + 6][n] = v1.z; Bs[kq + 7][n] = v1.w;
        }
        // Prefetch next A K-chunk into cache while this chunk is consumed
        if (kb + 32 < IN_C) {
            const float* pf = xb + (size_t)(kb + 32 + (tid >> 3)) * NPOS
                                 + m_base + (tid & 7) * 16;
            __builtin_prefetch(pf, 0, 3);
        }
        __syncthreads();

        #pragma unroll
        for (int ks = 0; ks < 8; ++ks) {
            const int k0 = ks * 4 + 2 * half;  // lanes 0-15: K={k,k+1}; 16-31: K={k+2,k+3}
            v2f a0, a1, b0, b1;
            a0.x = As[k0][mw + lm];       a0.y = As[k0 + 1][mw + lm];
            a1.x = As[k0][mw + 16 + lm];  a1.y = As[k0 + 1][mw + 16 + lm];
            b0.x = Bs[k0][nw + lm];       b0.y = Bs[k0 + 1][nw + lm];
            b1.x = Bs[k0][nw + 16 + lm];  b1.y = Bs[k0 + 1][nw + 16 + lm];
            acc00 = __builtin_amdgcn_wmma_f32_16x16x4_f32(
                        false, a0, false, b0, (short)0, acc00, false, false);
            acc01 = __builtin_amdgcn_wmma_f32_16x16x4_f32(
                        false, a0, false, b1, (short)0, acc01, false, false);
            acc10 = __builtin_amdgcn_wmma_f32_16x16x4_f32(
                        false, a1, false, b0, (short)0, acc10, false, false);
            acc11 = __builtin_amdgcn_wmma_f32_16x16x4_f32(
                        false, a1, false, b1, (short)0, acc11, false, false);
        }
        __syncthreads();
    }

    // Epilogue: C/D layout — VGPR v holds row M=v (lanes 0-15) / M=v+8 (lanes 16-31)
    const int out0  = o_base + nw + lm;
    const int out1  = out0 + 16;
    const float bias0 = conv_b[out0];
    const float bias1 = conv_b[out1];
    const int pos0 = m_base + mw + 8 * half;        // tile row base (i = 0)
    #pragma unroll
    for (int v = 0; v < 8; ++v) {
        size_t row0 = ((size_t)b * NPOS + pos0 + v) * EMBED;
        size_t row1 = ((size_t)b * NPOS + pos0 + 16 + v) * EMBED;
        P[row0 + out0] = acc00[v] + bias0;
        P[row0 + out1] = acc01[v] + bias1;
        P[row1 + out0] = acc10[v] + bias0;
        P[row1 + out1] = acc11[v] + bias1;
    }
}

// ---------------------------------------------------------------------------
// Kernel 3: one wave32 per (b,n) row of P: mu/var + s.p  ->  raw score
// ---------------------------------------------------------------------------
__global__ __launch_bounds__(256) void score_kernel(
    const float* __restrict__ P, const float* __restrict__ Sb,
    const float* __restrict__ scal, float* __restrict__ scores)
{
    const int row  = blockIdx.x * 8 + (threadIdx.x >> 5);  // 0..32767
    const int lane = threadIdx.x & 31;
    const int b = row >> 10;
    const float* p = P  + (size_t)row * EMBED;
    const float* s = Sb + b * EMBED;

    float sp = 0.f, spp = 0.f, sd = 0.f;
    for (int i = lane; i < EMBED; i += 32) {
        float v = p[i];
        sp += v; spp += v * v; sd += v * s[i];
    }
    #pragma unroll
    for (int off = 16; off > 0; off >>= 1) {
        sp  += __shfl_xor(sp,  off, 32);
        spp += __shfl_xor(spp, off, 32);
        sd  += __shfl_xor(sd,  off, 32);
    }
    if (lane == 0) {
        float mu   = sp  * (1.f / EMBED);
        float var  = spp * (1.f / EMBED) - mu * mu;
        float rstd = rsqrtf(var + 1e-5f);
        float sum_s = scal[b * 4 + 0];
        float rlnb  = scal[b * 4 + 1];
        float gterm = scal[b * 4 + 2];
        float cst   = scal[b * 4 + 3];
        scores[row] = (rstd * (sd - mu * sum_s) + rlnb + gterm + cst) * SCALE;
    }
}

// ---------------------------------------------------------------------------
// Kernel 4: per-batch softmax over 1024 grid scores -> d_out
// ---------------------------------------------------------------------------
__global__ __launch_bounds__(256) void softmax_kernel(
    const float* __restrict__ scores, float* __restrict__ out)
{
    const int b = blockIdx.x;
    const int tid = threadIdx.x;
    __shared__ float red[256];
    __shared__ float bmax, bsum;

    const float4 v = ((const float4*)(scores + b * NPOS))[tid];
    float m = fmaxf(fmaxf(v.x, v.y), fmaxf(v.z, v.w));
    red[tid] = m;
    __syncthreads();
    for (int off = 128; off > 0; off >>= 1) {
        if (tid < off) red[tid] = fmaxf(red[tid], red[tid + off]);
        __syncthreads();
    }
    if (tid == 0) bmax = red[0];
    __syncthreads();

    float e0 = expf(v.x - bmax), e1 = expf(v.y - bmax);
    float e2 = expf(v.z - bmax), e3 = expf(v.w - bmax);
    red[tid] = e0 + e1 + e2 + e3;
    __syncthreads();
    for (int off = 128; off > 0; off >>= 1) {
        if (tid < off) red[tid] += red[tid + off];
        __syncthreads();
    }
    if (tid == 0) bsum = red[0];
    __syncthreads();

    const float inv = 1.f / bsum;
    float4 o; o.x = e0 * inv; o.y = e1 * inv; o.z = e2 * inv; o.w = e3 * inv;
    ((float4*)out)[b * 256 + tid] = o;
}

// ---------------------------------------------------------------------------
extern "C" void kernel_launch(void* const* d_in, const int* in_sizes, int n_in,
                              void* d_out, int out_size, void* d_ws, size_t ws_size,
                              hipStream_t stream)
{
    const float* x      = (const float*)d_in[0];  // (32,1024,32,32)
    const float* gender = (const float*)d_in[1];  // (32,32)
    const float* conv_w = (const float*)d_in[2];  // (768,1024)
    const float* conv_b = (const float*)d_in[3];  // (768)
    const float* ln_g   = (const float*)d_in[4];  // (768)
    const float* ln_b   = (const float*)d_in[5];  // (768)
    const float* cls    = (const float*)d_in[6];  // (1,1,768)
    const float* qkv_w  = (const float*)d_in[7];  // (2400,800)
    const float* qkv_b  = (const float*)d_in[8];  // (2400)

    float* ws = (float*)d_ws;
    float* P      = ws;                                   // 32*1024*768
    float* Sb     = P + (size_t)BATCH * NPOS * EMBED;     // 32*768
    float* scal   = Sb + BATCH * EMBED;                   // 32*4
    float* scores = scal + BATCH * 4;                     // 32*1024

    prep_kernel<<<BATCH, 256, 0, stream>>>(gender, cls, qkv_w, qkv_b, ln_g, ln_b, Sb, scal);

    dim3 gg(NPOS / 128, EMBED / 64, BATCH);
    gemm_conv_kernel<<<gg, 256, 0, stream>>>(x, conv_w, conv_b, P);

    score_kernel<<<(BATCH * NPOS) / 8, 256, 0, stream>>>(P, Sb, scal, scores);

    softmax_kernel<<<BATCH, 256, 0, stream>>>(scores, (float*)d_out);
}